// LightGCN_22385369546775
// MI455X (gfx1250) — compile-verified
//
#include <hip/hip_runtime.h>

#define NUM_USERS 100000
#define NUM_ITEMS 50000
#define N_TOTAL   150000
#define LATENT    64
#define N_LAYERS  3

typedef __attribute__((ext_vector_type(2))) float v2f;
typedef __attribute__((ext_vector_type(8))) float v8f;

// ---------------------------------------------------------------------------
// init: acc = emb_in = concat(user_emb, item_emb), float4-vectorized
// ---------------------------------------------------------------------------
__global__ void lgcn_init(const float4* __restrict__ ue, const float4* __restrict__ ie,
                          float4* __restrict__ acc, float4* __restrict__ emb) {
    size_t i = (size_t)blockIdx.x * blockDim.x + threadIdx.x;
    const size_t nu = (size_t)NUM_USERS * LATENT / 4;
    const size_t nt = (size_t)N_TOTAL * LATENT / 4;
    if (i >= nt) return;
    float4 v = (i < nu) ? ue[i] : ie[i - nu];
    acc[i] = v;
    emb[i] = v;
}

__global__ void lgcn_zero(float4* __restrict__ p, size_t n4) {
    size_t i = (size_t)blockIdx.x * blockDim.x + threadIdx.x;
    if (i < n4) p[i] = make_float4(0.f, 0.f, 0.f, 0.f);
}

// ---------------------------------------------------------------------------
// COO SpMM scatter: 16 lanes per nonzero, float4 gather + 4x f32 atomic adds.
// Working set (~77MB ping-pong) sits in the 192MB L2, so both the random
// gather and the atomic scatter are L2-resident.
// ---------------------------------------------------------------------------
__global__ void lgcn_spmm(const float* __restrict__ vals, const int* __restrict__ rows,
                          const int* __restrict__ cols, const float* __restrict__ xin,
                          float* __restrict__ xout, int nnz) {
    size_t t = (size_t)blockIdx.x * blockDim.x + threadIdx.x;
    int e = (int)(t >> 4);               // nonzero index (16 threads per nnz)
    if (e >= nnz) return;
    int d4 = (int)(t & 15) * 4;          // 4 dims per thread
    int row = rows[e];
    int col = cols[e];
    float v  = vals[e];
    const float4 x = *(const float4*)(xin + (size_t)col * LATENT + d4);
    float* dst = xout + (size_t)row * LATENT + d4;
    atomicAdd(dst + 0, v * x.x);
    atomicAdd(dst + 1, v * x.y);
    atomicAdd(dst + 2, v * x.z);
    atomicAdd(dst + 3, v * x.w);
}

__global__ void lgcn_accum(float4* __restrict__ acc, const float4* __restrict__ emb, size_t n4) {
    size_t i = (size_t)blockIdx.x * blockDim.x + threadIdx.x;
    if (i >= n4) return;
    float4 a = acc[i];
    float4 e = emb[i];
    a.x += e.x; a.y += e.y; a.z += e.z; a.w += e.w;
    acc[i] = a;
}

// ---------------------------------------------------------------------------
// Batched dot products via V_WMMA_F32_16X16X4_F32.
// One wave computes a 16x16 tile C = U * I^T (K=64 via 16 chained WMMAs);
// gamma for the 16 pairs is the diagonal of C, scaled by 1/16 ((1/4)^2).
//
// A (16x4 f32) fragment: lane L holds U[L%16][2*(L/16) + {0,1}] for each
// K-chunk; B (4x16 f32) mirrors it with N = L%16 — so both fragments are one
// aligned float2 load per lane per chunk.
// ---------------------------------------------------------------------------
__global__ void lgcn_dot(const float* __restrict__ acc, const int* __restrict__ users,
                         const int* __restrict__ items, float* __restrict__ out) {
    int tile = blockIdx.x * (blockDim.x >> 5) + (threadIdx.x >> 5);
    int lane = threadIdx.x & 31;
    int half = lane >> 4;        // 0: K offsets {0,1};  1: K offsets {2,3}
    int l16  = lane & 15;        // M for A, N for B

    int u  = users[tile * 16 + l16];
    int it = items[tile * 16 + l16];
    const float* urow = acc + (size_t)u * LATENT;
    const float* irow = acc + (size_t)(NUM_USERS + it) * LATENT;

    v8f c = {0.f, 0.f, 0.f, 0.f, 0.f, 0.f, 0.f, 0.f};
#pragma unroll
    for (int kk = 0; kk < 16; ++kk) {
        int k = kk * 4 + half * 2;                 // even -> 8B aligned
        v2f a = *(const v2f*)(urow + k);
        v2f b = *(const v2f*)(irow + k);
        // D = A*B + C  (16x16x4 f32 WMMA, wave32)
        c = __builtin_amdgcn_wmma_f32_16x16x4_f32(
                /*neg_a=*/false, a, /*neg_b=*/false, b,
                /*c_mod=*/(short)0, c, /*reuse_a=*/false, /*reuse_b=*/false);
    }

    // Diagonal extraction: (m,m) at lane m / vgpr m (m<8), lane m+16 / vgpr m-8.
    const float scale = 1.0f / 16.0f;
    int r = -1, m = 0;
    if (lane < 8)        { r = lane;      m = lane;      }
    else if (lane >= 24) { r = lane - 24; m = lane - 16; }
    if (r >= 0) {
        float g = 0.f;
#pragma unroll
        for (int j = 0; j < 8; ++j)
            if (r == j) g = c[j];
        out[tile * 16 + m] = g * scale;
    }
}

// ---------------------------------------------------------------------------
extern "C" void kernel_launch(void* const* d_in, const int* in_sizes, int n_in,
                              void* d_out, int out_size, void* d_ws, size_t ws_size,
                              hipStream_t stream) {
    const float* ue   = (const float*)d_in[0];
    const float* ie   = (const float*)d_in[1];
    const float* vals = (const float*)d_in[2];
    const int*   rows = (const int*)d_in[3];
    const int*   cols = (const int*)d_in[4];
    const int*   users = (const int*)d_in[5];
    const int*   items = (const int*)d_in[6];
    float* out = (float*)d_out;

    const int nnz   = in_sizes[2];
    const int batch = in_sizes[5];

    // Workspace layout: [acc | embA | embB], each N_TOTAL*64 f32 (38.4 MB).
    const size_t S = (size_t)N_TOTAL * LATENT * sizeof(float);
    float* acc  = (float*)d_ws;
    float* embA = (float*)((char*)d_ws + S);
    float* embB = (float*)((char*)d_ws + 2 * S);

    const size_t n4 = (size_t)N_TOTAL * LATENT / 4;
    const int TB = 256;
    const int gElem = (int)((n4 + TB - 1) / TB);

    lgcn_init<<<gElem, TB, 0, stream>>>((const float4*)ue, (const float4*)ie,
                                        (float4*)acc, (float4*)embA);

    float* cur = embA;
    float* nxt = embB;
    const int gSpmm = (int)(((size_t)nnz * 16 + TB - 1) / TB);
    for (int layer = 0; layer < N_LAYERS; ++layer) {
        lgcn_zero<<<gElem, TB, 0, stream>>>((float4*)nxt, n4);
        lgcn_spmm<<<gSpmm, TB, 0, stream>>>(vals, rows, cols, cur, nxt, nnz);
        lgcn_accum<<<gElem, TB, 0, stream>>>((float4*)acc, (const float4*)nxt, n4);
        float* t = cur; cur = nxt; nxt = t;
    }

    // 16 pairs per wave, 8 waves per block -> exact full waves (EXEC all-ones).
    const int tiles = batch / 16;            // 1024
    lgcn_dot<<<tiles / 8, 256, 0, stream>>>(acc, users, items, out);
}